// DGCNN_8839042695334
// MI455X (gfx1250) — compile-verified
//
#include <hip/hip_runtime.h>
#include <hip/hip_bf16.h>
#include <math.h>

// ---- problem constants (match reference) ----
#define B_      32
#define N_      1024
#define M_      512        // N / POOL_RATIO
#define PATCH_  40
#define CH      16         // EQ * C_IN
#define COUT    16
#define NEG_SLOPE_ 0.2f
#define VN_EPS_    1e-6f

typedef float v2f __attribute__((ext_vector_type(2)));
typedef float v8f __attribute__((ext_vector_type(8)));

// ===================================================================
// Kernel 1: build y (B,N,16), ynorm, t (B,M,16), tnorm
// ===================================================================
__global__ __launch_bounds__(256)
void prep_kernel(const float* __restrict__ x0, const float* __restrict__ x1,
                 float* __restrict__ y, float* __restrict__ ynorm,
                 float* __restrict__ t, float* __restrict__ tnorm) {
  int gid = blockIdx.x * blockDim.x + threadIdx.x;
  if (gid >= B_ * M_) return;
  int b = gid / M_, m = gid % M_;
  int n0 = 2 * m, n1 = 2 * m + 1;
  float r0[CH], r1[CH];
#pragma unroll
  for (int c = 0; c < 4; ++c) {
    r0[c] = x0[(b * N_ + n0) * 4 + c];
    r1[c] = x0[(b * N_ + n1) * 4 + c];
  }
#pragma unroll
  for (int c = 0; c < 12; ++c) {
    r0[4 + c] = x1[(b * N_ + n0) * 12 + c];
    r1[4 + c] = x1[(b * N_ + n1) * 12 + c];
  }
  float s0 = 0.f, s1 = 0.f, st = 0.f;
#pragma unroll
  for (int c = 0; c < CH; ++c) {
    float tv = 0.5f * (r0[c] + r1[c]);
    y[(b * N_ + n0) * CH + c] = r0[c];
    y[(b * N_ + n1) * CH + c] = r1[c];
    t[(b * M_ + m) * CH + c] = tv;
    s0 += r0[c] * r0[c];
    s1 += r1[c] * r1[c];
    st += tv * tv;
  }
  ynorm[b * N_ + n0] = s0;
  ynorm[b * N_ + n1] = s1;
  tnorm[b * M_ + m] = st;
}

// ===================================================================
// Kernel 2: distance GEMM via V_WMMA_F32_16X16X4_F32 + stable top-40
// grid = B * (M/16) blocks, 128 threads (4 waves). Each wave computes
// a 16(M) x 256(N) strip of G = T * Y^T with 4 chained K=4 WMMAs per
// 16x16 tile; d2 tile lives in 64KB LDS; threads 0..15 then run a
// stable insertion top-40 per row (matches jax.lax.top_k ordering).
// ===================================================================
__global__ __launch_bounds__(128)
void dist_topk_kernel(const float* __restrict__ y, const float* __restrict__ ynorm,
                      const float* __restrict__ t, const float* __restrict__ tnorm,
                      int* __restrict__ idxout) {
  __shared__ float sd2[16][N_];       // 64 KB
  int b  = blockIdx.x >> 5;           // 32 blocks of 16 rows per batch
  int m0 = (blockIdx.x & 31) << 4;
  int tid  = threadIdx.x;
  int wave = tid >> 5;
  int lane = tid & 31;
  int half = lane >> 4;               // K-half selector for A/B fragments
  int l    = lane & 15;

  // A fragment: rows m0+l of T; ISA 16x4 f32 layout:
  // lanes 0-15: VGPR0=K0,VGPR1=K1 ; lanes16-31: VGPR0=K2,VGPR1=K3
  const float* Trow = &t[(b * M_ + m0 + l) * CH];
  v2f a[4];
#pragma unroll
  for (int kk = 0; kk < 4; ++kk) {
    a[kk].x = Trow[4 * kk + 2 * half];
    a[kk].y = Trow[4 * kk + 2 * half + 1];
  }
  // tnorm per D-row (D: vgpr r -> row r + 8*half)
  float tnr[8];
#pragma unroll
  for (int r = 0; r < 8; ++r) tnr[r] = tnorm[b * M_ + m0 + r + 8 * half];

  for (int i = 0; i < 16; ++i) {
    int n0c = (wave << 8) + (i << 4);
    // B fragment: B[k][n] = Y[n0c+n][k]; same per-lane gather as A
    const float* Yrow = &y[(b * N_ + n0c + l) * CH];
    v2f bf[4];
#pragma unroll
    for (int kk = 0; kk < 4; ++kk) {
      bf[kk].x = Yrow[4 * kk + 2 * half];
      bf[kk].y = Yrow[4 * kk + 2 * half + 1];
    }
    v8f acc = {0.f, 0.f, 0.f, 0.f, 0.f, 0.f, 0.f, 0.f};
#pragma unroll
    for (int kk = 0; kk < 4; ++kk) {
      acc = __builtin_amdgcn_wmma_f32_16x16x4_f32(
          /*neg_a=*/false, a[kk], /*neg_b=*/false, bf[kk],
          /*c_mod=*/(short)0, acc, /*reuse_a=*/false, /*reuse_b=*/false);
    }
    float yn = ynorm[b * N_ + n0c + l];
#pragma unroll
    for (int r = 0; r < 8; ++r) {
      // D layout: vgpr r, lanes0-15 -> M=r, N=lane ; lanes16-31 -> M=8+r
      sd2[r + 8 * half][n0c + l] = tnr[r] - 2.0f * acc[r] + yn;
    }
  }
  __syncthreads();

  if (tid < 16) {
    float bd[PATCH_];
    int   bi[PATCH_];
#pragma unroll
    for (int j = 0; j < PATCH_; ++j) { bd[j] = 3.402823466e38f; bi[j] = 0; }
    for (int n = 0; n < N_; ++n) {
      float v = sd2[tid][n];
      if (v < bd[PATCH_ - 1]) {
        int j = PATCH_ - 1;
        while (j > 0 && v < bd[j - 1]) {      // strict <: stable ties, lowest index first
          bd[j] = bd[j - 1]; bi[j] = bi[j - 1]; --j;
        }
        bd[j] = v; bi[j] = n;
      }
    }
    int m = m0 + tid;
    for (int j = 0; j < PATCH_; ++j)
      idxout[(b * M_ + m) * PATCH_ + j] = bi[j];
  }
}

// ===================================================================
// Kernel 3: batch-norm statistics. One thread per (b,m,patch) item:
// recompute p = edge @ featW, norm_c = sum_q p^2. Deterministic block
// reduction (shfl butterfly -> per-wave LDS slots -> fixed-order sum),
// per-block partial sums written to ws.
// ===================================================================
__global__ __launch_bounds__(256)
void vn_stats_kernel(const float* __restrict__ y, const float* __restrict__ t,
                     const int* __restrict__ idx,
                     const float* __restrict__ fW0, const float* __restrict__ fb0,
                     const float* __restrict__ fW1,
                     float* __restrict__ part) {
  __shared__ float sW0[128], sW1[128], sb0[16];
  __shared__ float swsum[8][16], swsq[8][16];
  int tid = threadIdx.x;
  if (tid < 128) { sW0[tid] = fW0[tid]; sW1[tid] = fW1[tid]; }
  if (tid < 16)  { sb0[tid] = fb0[tid]; }
  __syncthreads();

  int item = blockIdx.x * 256 + tid;   // grid sized exactly: B*M*PATCH = 2560*256
  int b   = item / (M_ * PATCH_);
  int rem = item % (M_ * PATCH_);
  int m   = rem / PATCH_;
  int k   = rem % PATCH_;
  int n   = idx[(b * M_ + m) * PATCH_ + k];
  const float* yr = &y[(b * N_ + n) * CH];
  const float* tr = &t[(b * M_ + m) * CH];

  float e0[8], e1[8], e2[8], e3[8];
#pragma unroll
  for (int j = 0; j < 8; ++j) {
    e0[j] = yr[j]     - 2.f * tr[j];       // (nb - t) - t
    e1[j] = yr[8 + j] - 2.f * tr[8 + j];
    e2[j] = tr[j];
    e3[j] = tr[8 + j];
  }
  float normc[16], sqc[16];
#pragma unroll
  for (int c = 0; c < 16; ++c) {
    float p0 = sb0[c], p1 = 0.f, p2 = 0.f, p3 = 0.f;
#pragma unroll
    for (int j = 0; j < 8; ++j) {
      p0 += e0[j] * sW0[j * 16 + c];
      p1 += e1[j] * sW1[j * 16 + c];
      p2 += e2[j] * sW1[j * 16 + c];
      p3 += e3[j] * sW1[j * 16 + c];
    }
    float nm = p0 * p0 + p1 * p1 + p2 * p2 + p3 * p3;
    normc[c] = nm;
    sqc[c]   = nm * nm;
  }
  // wave32 butterfly reduction (deterministic)
#pragma unroll
  for (int off = 16; off > 0; off >>= 1) {
#pragma unroll
    for (int c = 0; c < 16; ++c) {
      normc[c] += __shfl_xor(normc[c], off, 32);
      sqc[c]   += __shfl_xor(sqc[c],   off, 32);
    }
  }
  if ((tid & 31) == 0) {
    int w = tid >> 5;
#pragma unroll
    for (int c = 0; c < 16; ++c) { swsum[w][c] = normc[c]; swsq[w][c] = sqc[c]; }
  }
  __syncthreads();
  if (tid < 16) {
    float S = 0.f;
#pragma unroll
    for (int w = 0; w < 8; ++w) S += swsum[w][tid];
    part[blockIdx.x * 32 + tid] = S;
  } else if (tid < 32) {
    int c = tid - 16;
    float S = 0.f;
#pragma unroll
    for (int w = 0; w < 8; ++w) S += swsq[w][c];
    part[blockIdx.x * 32 + tid] = S;
  }
}

// ===================================================================
// Kernel 4: finalize BN stats (deterministic f64 serial reduction),
// produce per-channel affine: norm_bn = A*norm + Bc
// ===================================================================
__global__ void vn_finalize_kernel(const float* __restrict__ part,
                                   const float* __restrict__ gamma,
                                   const float* __restrict__ beta,
                                   float* __restrict__ ab) {
  int c = threadIdx.x;
  if (c >= 16) return;
  double S = 0.0, S2 = 0.0;
  for (int i = 0; i < 2560; ++i) {
    S  += (double)part[i * 32 + c];
    S2 += (double)part[i * 32 + 16 + c];
  }
  const double cnt = (double)(B_ * (long)M_ * PATCH_);
  double mu  = S / cnt;
  double var = S2 / cnt - mu * mu;
  double inv = 1.0 / sqrt(var + 1.0e-3);   // BN_EPS
  double g = (double)gamma[c];
  ab[c]      = (float)(g * inv);
  ab[16 + c] = (float)((double)beta[c] - g * mu * inv);
}

// ===================================================================
// Kernel 5: full per-patch pipeline + per-channel argmax + gather.
// One block (64 threads = 4 q-slots x 16 channels) per (b,m).
// ===================================================================
__global__ __launch_bounds__(64)
void out_kernel(const float* __restrict__ y, const float* __restrict__ t,
                const int* __restrict__ idx,
                const float* __restrict__ fW0, const float* __restrict__ fb0,
                const float* __restrict__ fW1,
                const float* __restrict__ dW0, const float* __restrict__ db0,
                const float* __restrict__ dW1,
                const float* __restrict__ pW0, const float* __restrict__ pb0,
                const float* __restrict__ pW1,
                const float* __restrict__ ab,
                float* __restrict__ out) {
  __shared__ float sFW0[128], sFW1[128], sFb[16];
  __shared__ float sDW0[128], sDW1[128], sDb[16];
  __shared__ float sPW0[256], sPW1[256], sPb[16];
  __shared__ float sA[16], sBc[16], st[16], snb[16];
  __shared__ int   sidx[PATCH_];
  __shared__ float red[3][4][16];
  __shared__ float syv[4][16];

  int tid = threadIdx.x;
  int bm = blockIdx.x;
  int b = bm / M_, m = bm % M_;

  for (int i = tid; i < 128; i += 64) {
    sFW0[i] = fW0[i]; sFW1[i] = fW1[i];
    sDW0[i] = dW0[i]; sDW1[i] = dW1[i];
  }
  for (int i = tid; i < 256; i += 64) { sPW0[i] = pW0[i]; sPW1[i] = pW1[i]; }
  if (tid < 16) {
    sFb[tid] = fb0[tid]; sDb[tid] = db0[tid]; sPb[tid] = pb0[tid];
    sA[tid] = ab[tid]; sBc[tid] = ab[16 + tid];
    st[tid] = t[(b * M_ + m) * CH + tid];
  }
  if (tid < PATCH_) sidx[tid] = idx[(b * M_ + m) * PATCH_ + tid];
  __syncthreads();

  int q = tid >> 4, c = tid & 15;
  float best = -3.402823466e38f;
  float outv = 0.f;

  for (int k = 0; k < PATCH_; ++k) {
    __syncthreads();                                    // protect snb/red reuse
    if (tid < 16) snb[tid] = y[(b * N_ + sidx[k]) * CH + tid];
    __syncthreads();

    float e[8];
    if (q == 0) {
#pragma unroll
      for (int j = 0; j < 8; ++j) e[j] = snb[j] - 2.f * st[j];
    } else if (q == 1) {
#pragma unroll
      for (int j = 0; j < 8; ++j) e[j] = snb[8 + j] - 2.f * st[8 + j];
    } else if (q == 2) {
#pragma unroll
      for (int j = 0; j < 8; ++j) e[j] = st[j];
    } else {
#pragma unroll
      for (int j = 0; j < 8; ++j) e[j] = st[8 + j];
    }
    const float* FW = (q == 0) ? sFW0 : sFW1;
    const float* DW = (q == 0) ? sDW0 : sDW1;
    float p = (q == 0) ? sFb[c] : 0.f;
    float d = (q == 0) ? sDb[c] : 0.f;
#pragma unroll
    for (int j = 0; j < 8; ++j) {
      p += e[j] * FW[j * 16 + c];
      d += e[j] * DW[j * 16 + c];
    }
    red[0][q][c] = p * p;
    red[1][q][c] = p * d;
    red[2][q][c] = d * d;
    __syncthreads();
    float norm = 0.f, dotpd = 0.f, dnorm = 0.f;
#pragma unroll
    for (int qq = 0; qq < 4; ++qq) {
      norm  += red[0][qq][c];
      dotpd += red[1][qq][c];
      dnorm += red[2][qq][c];
    }
    float scale = (sA[c] * norm + sBc[c]) / (norm + VN_EPS_);
    float ps  = p * scale;
    float dot = dotpd * scale;
    float yv  = (dot >= 0.f) ? ps : (ps - dot / (dnorm + VN_EPS_) * d);
    yv = NEG_SLOPE_ * ps + (1.f - NEG_SLOPE_) * yv;
    syv[q][c] = yv;
    __syncthreads();

    const float* PW = (q == 0) ? sPW0 : sPW1;
    float dp = (q == 0) ? sPb[c] : 0.f;
#pragma unroll
    for (int cc = 0; cc < 16; ++cc) dp += syv[q][cc] * PW[cc * 16 + c];
    red[0][q][c] = yv * dp;
    __syncthreads();
    float dots = red[0][0][c] + red[0][1][c] + red[0][2][c] + red[0][3][c];
    if (dots > best) { best = dots; outv = yv; }        // strict >: first max wins
  }
  out[((b * M_ + m) * 4 + q) * CH + c] = outv;
}

// ===================================================================
// Launch
// ===================================================================
extern "C" void kernel_launch(void* const* d_in, const int* in_sizes, int n_in,
                              void* d_out, int out_size, void* d_ws, size_t ws_size,
                              hipStream_t stream) {
  (void)in_sizes; (void)n_in; (void)out_size; (void)ws_size;
  const float* x0  = (const float*)d_in[0];
  const float* x1  = (const float*)d_in[1];
  const float* fW0 = (const float*)d_in[2];
  const float* fb0 = (const float*)d_in[3];
  const float* fW1 = (const float*)d_in[4];
  const float* dW0 = (const float*)d_in[5];
  const float* db0 = (const float*)d_in[6];
  const float* dW1 = (const float*)d_in[7];
  const float* bng = (const float*)d_in[8];
  const float* bnb = (const float*)d_in[9];
  const float* pW0 = (const float*)d_in[10];
  const float* pb0 = (const float*)d_in[11];
  const float* pW1 = (const float*)d_in[12];

  char* ws = (char*)d_ws;
  // workspace layout (bytes)
  float* y     = (float*)(ws + 0);                 // B*N*16  = 2,097,152 B
  float* ynorm = (float*)(ws + 2097152);           // B*N     =   131,072 B
  float* t     = (float*)(ws + 2228224);           // B*M*16  = 1,048,576 B
  float* tnorm = (float*)(ws + 3276800);           // B*M     =    65,536 B
  int*   idx   = (int*)  (ws + 3342336);           // B*M*40  = 2,621,440 B
  float* part  = (float*)(ws + 5963776);           // 2560*32 =   327,680 B
  float* ab    = (float*)(ws + 6291456);           // 32 floats

  prep_kernel<<<(B_ * M_ + 255) / 256, 256, 0, stream>>>(x0, x1, y, ynorm, t, tnorm);
  dist_topk_kernel<<<B_ * (M_ / 16), 128, 0, stream>>>(y, ynorm, t, tnorm, idx);
  vn_stats_kernel<<<(B_ * M_ * PATCH_) / 256, 256, 0, stream>>>(y, t, idx, fW0, fb0, fW1, part);
  vn_finalize_kernel<<<1, 16, 0, stream>>>(part, bng, bnb, ab);
  out_kernel<<<B_ * M_, 64, 0, stream>>>(y, t, idx, fW0, fb0, fW1, dW0, db0, dW1,
                                         pW0, pb0, pW1, ab, (float*)d_out);
}